// GroundedSam2_38800734552779
// MI455X (gfx1250) — compile-verified
//
#include <hip/hip_runtime.h>
#include <stdint.h>

#define N 8192
#define NW (N / 32)            // 256 suppression-mask words per row
#define IOU_THR 0.5f

// ---------------------------------------------------------------------------
// CDNA5 async global->LDS path (ASYNCcnt), guarded so both toolchains compile.
// Builtin signature (probe-verified via diagnostics): (AS1 int*, AS3 int*, imm, imm).
// ---------------------------------------------------------------------------
#if defined(__gfx1250__) && __has_builtin(__builtin_amdgcn_global_load_async_to_lds_b32)
#define HAS_ASYNC_LDS 1
typedef __attribute__((address_space(1))) int gas_int;
typedef __attribute__((address_space(3))) int las_int;
#define ASYNC_LD_B32(gsrc, ldst) \
    __builtin_amdgcn_global_load_async_to_lds_b32((gas_int*)(gsrc), (las_int*)(ldst), 0, 0)
#else
#define HAS_ASYNC_LDS 0
#endif

#if defined(__gfx1250__) && __has_builtin(__builtin_amdgcn_s_wait_asynccnt)
#define WAIT_ASYNC() __builtin_amdgcn_s_wait_asynccnt(0)
#else
#define WAIT_ASYNC() asm volatile("s_wait_asynccnt 0" ::: "memory")
#endif

// Workspace layout (float-sized slots):
//   X1:[0,N) Y1:[N,2N) X2:[2N,3N) Y2:[3N,4N) AREA:[4N,5N) S:[5N,6N) C:[6N,7N)
//   SUP: 256 u32 at [7N, 7N+256)     MASK: N*NW u32 at [8N, ...)  (8 MB)

// ---------------------------------------------------------------------------
// Suppression-mask matrix kernel (listed first so its CDNA5 async-to-LDS ops
// appear in the disasm snippet). Block = 256 threads (8 wave32) covering a
// 32-row x 256-col tile. Column boxes AND row boxes staged into LDS with
// global_load_async_to_lds_b32; one s_wait_asynccnt 0 + barrier covers all.
// Thread t -> (row r0 + t>>3, mask word t&7), builds one 32-bit word.
// ---------------------------------------------------------------------------
__global__ __launch_bounds__(256) void mask_kernel(
    const float* __restrict__ ws, unsigned* __restrict__ MASK)
{
    const float* X1 = ws;         const float* Y1 = ws + N;
    const float* X2 = ws + 2 * N; const float* Y2 = ws + 3 * N;
    const float* AR = ws + 4 * N;

    __shared__ float c_x1[256], c_y1[256], c_x2[256], c_y2[256], c_a[256];
    __shared__ float r_x1[32],  r_y1[32],  r_x2[32],  r_y2[32],  r_a[32];
    const int t  = threadIdx.x;
    const int c0 = blockIdx.x * 256;
    const int r0 = blockIdx.y * 32;

#if HAS_ASYNC_LDS
    ASYNC_LD_B32(X1 + c0 + t, &c_x1[t]);
    ASYNC_LD_B32(Y1 + c0 + t, &c_y1[t]);
    ASYNC_LD_B32(X2 + c0 + t, &c_x2[t]);
    ASYNC_LD_B32(Y2 + c0 + t, &c_y2[t]);
    ASYNC_LD_B32(AR + c0 + t, &c_a[t]);
    if (t < 32) {   // EXEC-masked async loads are legal (per-lane VMEM)
        ASYNC_LD_B32(X1 + r0 + t, &r_x1[t]);
        ASYNC_LD_B32(Y1 + r0 + t, &r_y1[t]);
        ASYNC_LD_B32(X2 + r0 + t, &r_x2[t]);
        ASYNC_LD_B32(Y2 + r0 + t, &r_y2[t]);
        ASYNC_LD_B32(AR + r0 + t, &r_a[t]);
    }
    WAIT_ASYNC();
#else
    c_x1[t] = X1[c0 + t]; c_y1[t] = Y1[c0 + t];
    c_x2[t] = X2[c0 + t]; c_y2[t] = Y2[c0 + t];
    c_a[t]  = AR[c0 + t];
    if (t < 32) {
        r_x1[t] = X1[r0 + t]; r_y1[t] = Y1[r0 + t];
        r_x2[t] = X2[r0 + t]; r_y2[t] = Y2[r0 + t];
        r_a[t]  = AR[r0 + t];
    }
#endif
    __syncthreads();

    const int lr  = t >> 3;          // local row 0..31
    const int row = r0 + lr;
    const int w   = t & 7;
    const float rx1 = r_x1[lr], ry1 = r_y1[lr], rx2 = r_x2[lr], ry2 = r_y2[lr];
    const float ra  = r_a[lr];

    unsigned bits = 0u;
    const int base = w * 32;
#pragma unroll 8
    for (int b = 0; b < 32; ++b) {
        const int lc = base + b;
        const int c  = c0 + lc;
        float iw = fminf(rx2, c_x2[lc]) - fmaxf(rx1, c_x1[lc]);
        float ih = fminf(ry2, c_y2[lc]) - fmaxf(ry1, c_y1[lc]);
        iw = fmaxf(iw, 0.0f); ih = fmaxf(ih, 0.0f);
        const float inter = iw * ih;
        const bool sup = (c > row) && (inter > IOU_THR * (ra + c_a[lc] - inter));
        bits |= sup ? (1u << b) : 0u;
    }
    MASK[(size_t)row * NW + (c0 >> 5) + w] = bits;  // every word written exactly once
}

// ---------------------------------------------------------------------------
// Bitonic argsort by -score in LDS (one workgroup, 1024 threads),
// then emit sorted SoA box data + areas + scores + categories.
// ---------------------------------------------------------------------------
__global__ __launch_bounds__(1024) void sort_kernel(
    const float* __restrict__ boxes, const float* __restrict__ scores,
    const int* __restrict__ cats, float* __restrict__ ws)
{
    __shared__ float key[N];   // 32 KB
    __shared__ int   sidx[N];  // 32 KB  (64 KB total, well under 320 KB/WGP)
    const int tid = threadIdx.x;

    for (int i = tid; i < N; i += 1024) { key[i] = -scores[i]; sidx[i] = i; }
    __syncthreads();

    for (int k = 2; k <= N; k <<= 1) {
        for (int j = k >> 1; j > 0; j >>= 1) {
            for (int i = tid; i < N; i += 1024) {
                int ixj = i ^ j;
                if (ixj > i) {
                    bool up = ((i & k) == 0);
                    float a = key[i], b = key[ixj];
                    if ((a > b) == up) {
                        key[i] = b; key[ixj] = a;
                        int t2 = sidx[i]; sidx[i] = sidx[ixj]; sidx[ixj] = t2;
                    }
                }
            }
            __syncthreads();
        }
    }

    float* X1 = ws;          float* Y1 = ws + N;     float* X2 = ws + 2 * N;
    float* Y2 = ws + 3 * N;  float* AR = ws + 4 * N; float* S  = ws + 5 * N;
    int*   C  = (int*)(ws + 6 * N);
    for (int i = tid; i < N; i += 1024) {
        int o = sidx[i];
        float x1 = boxes[o * 4 + 0], y1 = boxes[o * 4 + 1];
        float x2 = boxes[o * 4 + 2], y2 = boxes[o * 4 + 3];
        X1[i] = x1; Y1[i] = y1; X2[i] = x2; Y2[i] = y2;
        AR[i] = (x2 - x1) * (y2 - y1);
        S[i]  = scores[o];
        C[i]  = cats[o];
    }
}

// ---------------------------------------------------------------------------
// Sequential greedy reduce. One 256-thread block; suppressed bitmask lives in
// LDS (256 words). Row i's bit i is never set by row i (j>i only), so the
// kept-test is race-free within an iteration; barrier orders iterations.
// Prefetch next mask row (global_prefetch_b8) to hide L2 latency.
// ---------------------------------------------------------------------------
__global__ __launch_bounds__(256) void reduce_kernel(
    const unsigned* __restrict__ MASK, unsigned* __restrict__ SUP)
{
    __shared__ unsigned sup[NW];
    const int t = threadIdx.x;
    sup[t] = 0u;
    __syncthreads();

    for (int i = 0; i < N; ++i) {
        const unsigned wv = sup[i >> 5];
        const bool kept = ((wv >> (i & 31)) & 1u) == 0u;
        if (i + 1 < N) __builtin_prefetch(&MASK[(size_t)(i + 1) * NW + t], 0, 0);
        if (kept) sup[t] |= MASK[(size_t)i * NW + t];
        __syncthreads();
    }
    SUP[t] = sup[t];
}

// ---------------------------------------------------------------------------
// Fallback (tiny workspace): single-block NMS directly from sorted SoA data.
// ---------------------------------------------------------------------------
__global__ __launch_bounds__(1024) void nms_serial_kernel(
    const float* __restrict__ ws, unsigned* __restrict__ SUP)
{
    const float* X1 = ws;         const float* Y1 = ws + N;
    const float* X2 = ws + 2 * N; const float* Y2 = ws + 3 * N;
    const float* AR = ws + 4 * N;

    __shared__ unsigned sup[NW];
    const int t = threadIdx.x;
    for (int k = t; k < NW; k += 1024) sup[k] = 0u;
    __syncthreads();

    for (int i = 0; i < N; ++i) {
        const bool kept = ((sup[i >> 5] >> (i & 31)) & 1u) == 0u;
        if (kept) {
            const float rx1 = X1[i], ry1 = Y1[i], rx2 = X2[i], ry2 = Y2[i], ra = AR[i];
            for (int j = i + 1 + t; j < N; j += 1024) {
                float iw = fminf(rx2, X2[j]) - fmaxf(rx1, X1[j]);
                float ih = fminf(ry2, Y2[j]) - fmaxf(ry1, Y1[j]);
                iw = fmaxf(iw, 0.0f); ih = fmaxf(ih, 0.0f);
                const float inter = iw * ih;
                if (inter > IOU_THR * (ra + AR[j] - inter))
                    atomicOr(&sup[j >> 5], 1u << (j & 31));
            }
        }
        __syncthreads();
    }
    for (int k = t; k < NW; k += 1024) SUP[k] = sup[k];
}

// ---------------------------------------------------------------------------
// Outputs: boxes*keep (4N), cats or -1 (N), scores*keep (N).
// ---------------------------------------------------------------------------
__global__ __launch_bounds__(256) void output_kernel(
    const float* __restrict__ ws, const unsigned* __restrict__ SUP,
    float* __restrict__ out)
{
    const int i = blockIdx.x * 256 + threadIdx.x;
    if (i >= N) return;
    const float* X1 = ws;         const float* Y1 = ws + N;
    const float* X2 = ws + 2 * N; const float* Y2 = ws + 3 * N;
    const float* S  = ws + 5 * N; const int*   C  = (const int*)(ws + 6 * N);

    const bool keep = ((SUP[i >> 5] >> (i & 31)) & 1u) == 0u;
    const float k = keep ? 1.0f : 0.0f;
    out[i * 4 + 0] = X1[i] * k;
    out[i * 4 + 1] = Y1[i] * k;
    out[i * 4 + 2] = X2[i] * k;
    out[i * 4 + 3] = Y2[i] * k;
    out[4 * N + i] = keep ? (float)C[i] : -1.0f;
    out[5 * N + i] = S[i] * k;
}

// ---------------------------------------------------------------------------
extern "C" void kernel_launch(void* const* d_in, const int* in_sizes, int n_in,
                              void* d_out, int out_size, void* d_ws, size_t ws_size,
                              hipStream_t stream)
{
    (void)in_sizes; (void)n_in; (void)out_size;
    const float* boxes  = (const float*)d_in[0];
    const float* scores = (const float*)d_in[1];
    const int*   cats   = (const int*)d_in[2];
    float*    ws   = (float*)d_ws;
    float*    out  = (float*)d_out;
    unsigned* SUP  = (unsigned*)(ws + 7 * N);
    unsigned* MASK = (unsigned*)(ws + 8 * N);

    sort_kernel<<<1, 1024, 0, stream>>>(boxes, scores, cats, ws);

    const size_t need = (size_t)8 * N * 4 + (size_t)N * NW * 4;  // ~8.6 MB
    if (ws_size >= need) {
        dim3 grid(N / 256, N / 32);  // 32 x 256 tiles
        mask_kernel<<<grid, 256, 0, stream>>>(ws, MASK);
        reduce_kernel<<<1, 256, 0, stream>>>(MASK, SUP);
    } else {
        nms_serial_kernel<<<1, 1024, 0, stream>>>(ws, SUP);
    }

    output_kernel<<<(N + 255) / 256, 256, 0, stream>>>(ws, SUP, out);
}